// AttDistance_1580547972197
// MI455X (gfx1250) — compile-verified
//
#include <hip/hip_runtime.h>
#include <math.h>

#define TQ   1024
#define TY   1024
#define DD   64
#define QBLK 16
#define NWAVE 8

typedef __attribute__((ext_vector_type(2))) float v2f;
typedef __attribute__((ext_vector_type(8))) float v8f;
typedef __attribute__((vector_size(16))) int v4i;

#define AS1 __attribute__((address_space(1)))
#define AS3 __attribute__((address_space(3)))

#if defined(__HIP_DEVICE_COMPILE__) && __has_builtin(__builtin_amdgcn_global_load_async_to_lds_b128)
#define USE_ASYNC 1
#else
#define USE_ASYNC 0
#endif

#if defined(__HIP_DEVICE_COMPILE__) && __has_builtin(__builtin_amdgcn_s_wait_asynccnt)
#define WAIT_ASYNC(n) do { __builtin_amdgcn_s_wait_asynccnt(n); asm volatile("" ::: "memory"); } while (0)
#else
#define WAIT_ASYNC(n) do { asm volatile("s_wait_asynccnt %0" :: "i"(n) : "memory"); } while (0)
#endif

// Workgroup: 256 threads = 8 waves. Covers 16 query rows x 1024 ty.
// Wave w owns ty in [128w, 128w+128). Lane = 16h + r: owns q-row r,
// ty_local = s*8 + 4h + j for subchunk s in [0,16), j in [0,4).
extern "C" __global__ void __launch_bounds__(256)
att_l1_softmax_kernel(const float* __restrict__ Q, const float* __restrict__ Y,
                      float* __restrict__ att, float* __restrict__ sim)
{
  __shared__ __align__(16) float yBuf[NWAVE][2][8 * DD];  // per-wave double buffer, 2KB each
  __shared__ float maxBuf[16][QBLK];
  __shared__ float sumBuf[16][QBLK];
  __shared__ float rowSumBuf[QBLK];

  const int tid  = threadIdx.x;
  const int w    = tid >> 5;
  const int lane = tid & 31;
  const int h    = lane >> 4;
  const int r    = lane & 15;

  const int b    = blockIdx.x >> 6;          // TQ/QBLK = 64 blocks per batch
  const int q0   = (blockIdx.x & 63) * QBLK;
  const int qRow = q0 + r;

  // ---- load this lane's query row (64 floats) into registers ----
  const float4* qp = (const float4*)(Q + ((size_t)b * TQ + qRow) * DD);
  float4 qv[16];
#pragma unroll
  for (int kc = 0; kc < 16; ++kc) qv[kc] = qp[kc];

  const int   tyW  = w * 128;
  const char* ySrc = (const char*)(Y + ((size_t)b * TY + tyW) * DD);
  char* yDst[2] = { (char*)&yBuf[w][0][0], (char*)&yBuf[w][1][0] };

  // copy subchunk s: 8 ty rows x 64 floats = 2048B -> per-wave LDS buffer (s&1)
  auto issue_copy = [&](int s) {
    const char* g = ySrc + (size_t)s * 8 * DD * sizeof(float);
    char*       l = yDst[s & 1];
#if USE_ASYNC
#pragma unroll
    for (int u = 0; u < 4; ++u) {
      const int off = (lane + 32 * u) * 16;  // 128 x 16B chunks across 32 lanes
      __builtin_amdgcn_global_load_async_to_lds_b128(
          (AS1 v4i*)(g + off), (AS3 v4i*)(l + off), 0, 0);
    }
#else
#pragma unroll
    for (int u = 0; u < 4; ++u) {
      const int off = (lane + 32 * u) * 16;
      *(float4*)(l + off) = *(const float4*)(g + off);
    }
#endif
  };

  float dv[64];
  float lmax = -3.4e38f;

  issue_copy(0);
#pragma unroll
  for (int s = 0; s < 16; ++s) {
#if USE_ASYNC
    if (s < 15) { issue_copy(s + 1); WAIT_ASYNC(4); }
    else        { WAIT_ASYNC(0); }
#else
    __syncthreads();   // make copy(s) visible across lanes of the wave
#endif
    // ---- distance: all 16 lanes of a half process the same ty -> LDS broadcast ----
#pragma unroll
    for (int j = 0; j < 4; ++j) {
      const float4* yc = (const float4*)&yBuf[w][s & 1][(h * 4 + j) * DD];
      float acc = 0.f;
#pragma unroll
      for (int kc = 0; kc < 16; ++kc) {
        const float4 yv = yc[kc];
        const float4 qq = qv[kc];
        acc += fabsf(qq.x - yv.x);
        acc += fabsf(qq.y - yv.y);
        acc += fabsf(qq.z - yv.z);
        acc += fabsf(qq.w - yv.w);
      }
      const float val = -acc * (1.0f / (float)DD);  // att pre-softmax = -mean|q-y|
      dv[s * 4 + j] = val;
      lmax = fmaxf(lmax, val);
    }
#if !USE_ASYNC
    __syncthreads();
    if (s < 15) issue_copy(s + 1);
#endif
  }

  // ---- row max across the 16 (wave,half) partials ----
  maxBuf[w * 2 + h][r] = lmax;
  __syncthreads();
  float rmax = -3.4e38f;
#pragma unroll
  for (int j = 0; j < 16; ++j) rmax = fmaxf(rmax, maxBuf[j][r]);

  if (tid < 16) sim[(size_t)b * TQ + q0 + tid] = rmax;   // tid<16 => w=0,h=0,r=tid

  // ---- exp + partial sum in registers ----
  float psum = 0.f;
#pragma unroll
  for (int i = 0; i < 64; ++i) {
    const float e = __expf(dv[i] - rmax);
    dv[i] = e;
    psum += e;
  }
  sumBuf[w * 2 + h][r] = psum;
  __syncthreads();

  // ---- cross-wave sum reduction on the matrix unit: ones(16x4) x partial(4x16), 4 chunks ----
  if (w == 0) {
#if defined(__HIP_DEVICE_COMPILE__) && __has_builtin(__builtin_amdgcn_wmma_f32_16x16x4_f32)
    v2f a; a.x = 1.0f; a.y = 1.0f;                 // A = all-ones 16x4
    v8f c = {0.f, 0.f, 0.f, 0.f, 0.f, 0.f, 0.f, 0.f};
#pragma unroll
    for (int c4 = 0; c4 < 4; ++c4) {
      v2f bm;                                      // B[k][n]: n=lane&15, k=2h+{0,1}
      bm.x = sumBuf[4 * c4 + 2 * h + 0][r];
      bm.y = sumBuf[4 * c4 + 2 * h + 1][r];
      c = __builtin_amdgcn_wmma_f32_16x16x4_f32(false, a, false, bm,
                                                (short)0, c, false, false);
    }
    if (lane < 16) rowSumBuf[lane] = c[0];         // C[0][n] = rowSum[n]
#else
    if (lane < 16) {
      float ssum = 0.f;
#pragma unroll
      for (int j = 0; j < 16; ++j) ssum += sumBuf[j][lane];
      rowSumBuf[lane] = ssum;
    }
#endif
  }
  __syncthreads();

  // ---- normalize and write att (each lane: 16 x float4, 16B-aligned) ----
  const float inv = 1.0f / rowSumBuf[r];
  float* attRow = att + ((size_t)b * TQ + qRow) * TY + tyW;
#pragma unroll
  for (int s = 0; s < 16; ++s) {
    float4 o;
    o.x = dv[s * 4 + 0] * inv;
    o.y = dv[s * 4 + 1] * inv;
    o.z = dv[s * 4 + 2] * inv;
    o.w = dv[s * 4 + 3] * inv;
    *(float4*)(attRow + s * 8 + h * 4) = o;
  }
}

extern "C" void kernel_launch(void* const* d_in, const int* in_sizes, int n_in,
                              void* d_out, int out_size, void* d_ws, size_t ws_size,
                              hipStream_t stream) {
  const float* Q = (const float*)d_in[0];
  const float* Y = (const float*)d_in[1];
  float* att = (float*)d_out;
  const int B = in_sizes[0] / (TQ * DD);
  float* sim = att + (size_t)B * TQ * TY;   // outputs concatenated: att then sim
  dim3 grid(B * (TQ / QBLK));               // 256 workgroups for B=4
  att_l1_softmax_kernel<<<grid, dim3(256), 0, stream>>>(Q, Y, att, sim);
}